// PixelwiseMLPRelight_80255758893162
// MI455X (gfx1250) — compile-verified
//
#include <hip/hip_runtime.h>

typedef float v2f __attribute__((ext_vector_type(2)));
typedef float v3f __attribute__((ext_vector_type(3), aligned(4)));
typedef float v4f __attribute__((ext_vector_type(4)));
typedef float v8f __attribute__((ext_vector_type(8)));

#define HW_PIX (1024 * 1024)
#define PIX_STRIDE 66            // even -> 8B-aligned v2f LDS reads; banks 2p%64 conflict-free
#define DUMP_OFF (32 * PIX_STRIDE)
#define WAVE_LDS (32 * PIX_STRIDE + 80)   // + dump row for lanes 16..31 (they hold exact zeros)
#define WAVES_PER_BLOCK 4
#define BLOCK (WAVES_PER_BLOCK * 32)

__device__ __forceinline__ float fast_sigmoid(float x) {
    // sigmoid(x) = 1 / (1 + 2^(-x*log2(e))) ; v_exp_f32 + v_rcp_f32 are ~1 ulp
    float e = __builtin_amdgcn_exp2f(x * -1.4426950408889634f);
    return __builtin_amdgcn_rcpf(1.0f + e);
}

__device__ __forceinline__ v2f make2(float x, float y) { v2f r; r.x = x; r.y = y; return r; }

__global__ __launch_bounds__(BLOCK) void relight_wmma_kernel(
    const float* __restrict__ xld,   // light_direction [8,2]
    const float* __restrict__ w1,    // [HW,8,2]
    const float* __restrict__ b1,    // [HW,8]
    const float* __restrict__ w2,    // [HW,3,8]
    const float* __restrict__ b2,    // [HW,3]
    float* __restrict__ out)         // [8,HW,3]
{
    __shared__ float lds[WAVES_PER_BLOCK * WAVE_LDS];

    const int tid  = threadIdx.x;
    const int lane = tid & 31;
    const int wave = tid >> 5;
    float* wlds = lds + wave * WAVE_LDS;

    const int pixbase = blockIdx.x * BLOCK + wave * 32;
    const int g  = (lane >> 3) & 1;   // which pixel of the pair this lane serves
    const int k7 = lane & 7;          // hidden-unit index within the pixel

    // ---- Prefetch all per-lane phase-2 operands first (overlaps all of phase 1).
    const size_t pix = (size_t)pixbase + lane;
    v2f w2p[3][4];
    {
        const v4f* wp = (const v4f*)(w2 + pix * 24);   // 96 B/pixel, 16B aligned
        #pragma unroll
        for (int o = 0; o < 3; ++o) {
            v4f t0 = __builtin_nontemporal_load(wp + 2 * o);
            v4f t1 = __builtin_nontemporal_load(wp + 2 * o + 1);
            w2p[o][0] = make2(t0.x, t0.y); w2p[o][1] = make2(t0.z, t0.w);
            w2p[o][2] = make2(t1.x, t1.y); w2p[o][3] = make2(t1.z, t1.w);
        }
    }
    v2f b1p[4];
    {
        const v4f* bp = (const v4f*)(b1 + pix * 8);    // 32 B/pixel, 16B aligned
        v4f t0 = __builtin_nontemporal_load(bp);
        v4f t1 = __builtin_nontemporal_load(bp + 1);
        b1p[0] = make2(t0.x, t0.y); b1p[1] = make2(t0.z, t0.w);
        b1p[2] = make2(t1.x, t1.y); b1p[3] = make2(t1.z, t1.w);
    }
    const v3f bb = __builtin_nontemporal_load((const v3f*)(b2 + pix * 3));

    // ---- A operand (16x4 f32): A[m=batch][k=d]; VGPR0={K0|K2}, VGPR1={K1|K3}.
    // Lanes 8..31 are exactly zero: K=2,3 columns kill the duplicated B rows,
    // and rows 8..15 of D become exactly 0 (C is 0), so lanes 16..31 are benign.
    v2f a; a.x = 0.0f; a.y = 0.0f;
    if (lane < 8) { a.x = xld[2 * lane]; a.y = xld[2 * lane + 1]; }

    // ---- Stage ALL W1 (B operand) loads: branchless, one burst, imm offsets.
    const float* pb = w1 + ((size_t)pixbase + g) * 16 + (size_t)k7 * 2;
    v2f breg[16];
    #pragma unroll
    for (int i = 0; i < 16; ++i)
        breg[i] = __builtin_nontemporal_load((const v2f*)(pb + 32 * i)); // +128 B/pair

    // ---- Branchless LDS-store addressing.
    // Lanes 0..15: pixel-pair rows, advancing 2 pixels (132 dwords) per WMMA.
    // Lanes 16..31: hold exact zeros; park them in the dump row (step 0).
    const int sbase = ((lane < 16) ? 0 : DUMP_OFF) + PIX_STRIDE * g + k7;
    const int sstep = (lane < 16) ? (2 * PIX_STRIDE) : 0;

    // ---- Phase 1: 16x V_WMMA_F32_16X16X4_F32, C = 0 (loop-invariant), raw store.
    const v8f czero = {0.f, 0.f, 0.f, 0.f, 0.f, 0.f, 0.f, 0.f};
    #pragma unroll
    for (int i = 0; i < 16; ++i) {
        v8f d = __builtin_amdgcn_wmma_f32_16x16x4_f32(
            /*neg_a=*/false, a, /*neg_b=*/false, breg[i],
            /*c_mod=*/(short)0, czero, /*reuse_a=*/false, /*reuse_b=*/false);
        float* q = wlds + sbase + i * sstep;
        #pragma unroll
        for (int v = 0; v < 8; ++v)
            q[v * 8] = d[v];                 // [pix][batch][k], merged ds_store_2addr
    }

    __syncthreads();   // make the cross-lane LDS handoff airtight

    // ---- Phase 2: each lane owns one pixel. bias+ReLU once per (batch,k),
    // packed-f32 FMA chain (v_pk_fma_f32 / v_pk_add_f32 on gfx1250).
    const v2f* hptr = (const v2f*)(wlds + (size_t)lane * PIX_STRIDE);  // 8B aligned
    const v2f zero2 = {0.f, 0.f};

    #pragma unroll
    for (int bi = 0; bi < 8; ++bi) {
        v2f acc0 = make2(bb.x, 0.f);
        v2f acc1 = make2(bb.y, 0.f);
        v2f acc2 = make2(bb.z, 0.f);
        #pragma unroll
        for (int j = 0; j < 4; ++j) {
            v2f h = hptr[bi * 4 + j];                            // ds_load_b64
            h = __builtin_elementwise_max(h + b1p[j], zero2);    // bias + ReLU
            acc0 = __builtin_elementwise_fma(h, w2p[0][j], acc0);
            acc1 = __builtin_elementwise_fma(h, w2p[1][j], acc1);
            acc2 = __builtin_elementwise_fma(h, w2p[2][j], acc2);
        }
        v3f o;
        o.x = fast_sigmoid(acc0.x + acc0.y);
        o.y = fast_sigmoid(acc1.x + acc1.y);
        o.z = fast_sigmoid(acc2.x + acc2.y);
        __builtin_nontemporal_store(o, (v3f*)(out + ((size_t)bi * HW_PIX + pix) * 3));
    }
}

extern "C" void kernel_launch(void* const* d_in, const int* in_sizes, int n_in,
                              void* d_out, int out_size, void* d_ws, size_t ws_size,
                              hipStream_t stream) {
    const float* xld = (const float*)d_in[0];   // light_direction [8,2]
    const float* w1  = (const float*)d_in[1];   // [1024,1024,8,2]
    const float* b1  = (const float*)d_in[2];   // [1024,1024,8]
    const float* w2  = (const float*)d_in[3];   // [1024,1024,3,8]
    const float* b2  = (const float*)d_in[4];   // [1024,1024,3]
    float* out = (float*)d_out;                 // [8,1024,1024,3]

    dim3 grid(HW_PIX / BLOCK);   // 8192 blocks
    dim3 block(BLOCK);           // 128 threads = 4 waves
    relight_wmma_kernel<<<grid, block, 0, stream>>>(xld, w1, b1, w2, b2, out);
}